// LossFunction_48945447306133
// MI455X (gfx1250) — compile-verified
//
#include <hip/hip_runtime.h>

typedef float f4  __attribute__((ext_vector_type(4)));
typedef float v2f __attribute__((ext_vector_type(2)));
typedef float v8f __attribute__((ext_vector_type(8)));

#define NBLOCKS  1024
#define NTHREADS 256

__device__ __forceinline__ float sl1(float x, float y) {
    float d = fabsf(x - y);
    return d < 1.0f ? 0.5f * d * d : d - 0.5f;
}

// Stage 1: stream both tensors once with NT hints, accumulate the 4 loss
// partial sums, block-reduce, write 4 floats per block into d_ws.
__global__ __launch_bounds__(NTHREADS)
void loss_partial_kernel(const float* __restrict__ t,
                         const float* __restrict__ p,
                         float* __restrict__ partials,
                         unsigned int numGroups) {
    float s1 = 0.f, s2 = 0.f, s3 = 0.f, s4 = 0.f;
    const unsigned int stride = gridDim.x * blockDim.x;

    for (unsigned int g = blockIdx.x * blockDim.x + threadIdx.x;
         g < numGroups; g += stride) {
        const f4* t4 = (const f4*)(t + (size_t)g * 52);  // 4 rows = 13 x float4, 16B aligned
        const f4* p4 = (const f4*)(p + (size_t)g * 52);

        if (g + stride < numGroups) {   // gfx1250 global_prefetch_b8
            __builtin_prefetch(t + (size_t)(g + stride) * 52, 0, 0);
            __builtin_prefetch(p + (size_t)(g + stride) * 52, 0, 0);
        }

        float tf[52], pf[52];
#pragma unroll
        for (int i = 0; i < 13; ++i) {
            f4 tv = __builtin_nontemporal_load(t4 + i);
            f4 pv = __builtin_nontemporal_load(p4 + i);
            tf[4*i+0] = tv.x; tf[4*i+1] = tv.y; tf[4*i+2] = tv.z; tf[4*i+3] = tv.w;
            pf[4*i+0] = pv.x; pf[4*i+1] = pv.y; pf[4*i+2] = pv.z; pf[4*i+3] = pv.w;
        }

#pragma unroll
        for (int r = 0; r < 4; ++r) {
            const int o = 13 * r;
            float t0 = tf[o+0], t1 = tf[o+1], t2 = tf[o+2], t3 = tf[o+3];
            float p0 = pf[o+0], p1 = pf[o+1], p2 = pf[o+2], p3 = pf[o+3];

            s2 += sl1(t0,p0) + sl1(t1,p1) + sl1(t2,p2) + sl1(t3,p3);

            float xx1 = fmaxf(t0, p0), yy1 = fmaxf(t1, p1);
            float xx2 = fminf(t2, p2), yy2 = fminf(t3, p3);
            float w = fmaxf(xx2 - xx1, 0.f), h = fmaxf(yy2 - yy1, 0.f);
            float inter = w * h;
            float a1 = (t2 - t0) * (t3 - t1);
            float a2 = (p2 - p0) * (p3 - p1);
            float iou = inter / (a1 + a2 - inter + 1e-7f);
            s1 += sl1(1.0f, iou);

#pragma unroll
            for (int f = 4; f < 12; ++f) s4 += sl1(tf[o+f], pf[o+f]);
            s3 += sl1(tf[o+12], pf[o+12]);
        }
    }

    __shared__ float m0[NTHREADS], m1[NTHREADS], m2[NTHREADS], m3[NTHREADS];
    const int tid = threadIdx.x;
    m0[tid] = s1; m1[tid] = s2; m2[tid] = s3; m3[tid] = s4;
    for (int s = NTHREADS / 2; s > 0; s >>= 1) {
        __syncthreads();
        if (tid < s) {
            m0[tid] += m0[tid+s]; m1[tid] += m1[tid+s];
            m2[tid] += m2[tid+s]; m3[tid] += m3[tid+s];
        }
    }
    if (tid == 0) {
        float* outp = partials + (size_t)blockIdx.x * 4;
        outp[0] = m0[0]; outp[1] = m1[0]; outp[2] = m2[0]; outp[3] = m3[0];
    }
}

// Stage 2: one wave32. Deterministic reduce of nblocks x 4 partials using
// V_WMMA_F32_16X16X4_F32: A(16x4) row m carries only component (m&3),
// B = ones(4x16), C accumulates row-sums. Per the CDNA5 C/D layout,
// D[m][0] lives in lane 0 (m=0..7 -> vgpr 0..7) and lane 16 (m=8..15).
__global__ __launch_bounds__(32)
void loss_final_kernel(const float* __restrict__ P,
                       float* __restrict__ out,
                       float invR, int nblocks) {
    const int L  = threadIdx.x;
    const int c  = L & 3;            // component carried by this A row
    const int mH = (L >> 2) & 3;     // row-group -> block sub-index
    const int k0 = (L >> 4) * 2;     // K pair held by this half-wave

    v2f b; b.x = 1.0f; b.y = 1.0f;   // ones matrix (layout-invariant)
    v8f acc = {};

    for (int base = 0; base < nblocks; base += 16) {   // 16 blocks / WMMA
        const int blk = base + mH * 4 + k0;
        v2f a;
        a.x = P[(blk    ) * 4 + c];
        a.y = P[(blk + 1) * 4 + c];
        acc = __builtin_amdgcn_wmma_f32_16x16x4_f32(
            /*neg_a=*/false, a, /*neg_b=*/false, b,
            /*c_mod=*/(short)0, acc, /*reuse_a=*/false, /*reuse_b=*/false);
    }

    // component c total = D[c][0] + D[c+4][0] + D[c+8][0] + D[c+12][0]
    float tc0 = acc[0] + acc[4];
    float tc1 = acc[1] + acc[5];
    float tc2 = acc[2] + acc[6];
    float tc3 = acc[3] + acc[7];

    float s1 = __shfl(tc0, 0, 32) + __shfl(tc0, 16, 32);
    float s2 = __shfl(tc1, 0, 32) + __shfl(tc1, 16, 32);
    float s3 = __shfl(tc2, 0, 32) + __shfl(tc2, 16, 32);
    float s4 = __shfl(tc3, 0, 32) + __shfl(tc3, 16, 32);

    if (L == 0) {
        // loss1/R + loss2/(4R) + loss3/R + 0.5*loss4/(8R)
        out[0] = s1 * invR + s2 * (0.25f * invR)
               + s3 * invR + s4 * (0.0625f * invR);
    }
}

extern "C" void kernel_launch(void* const* d_in, const int* in_sizes, int n_in,
                              void* d_out, int out_size, void* d_ws, size_t ws_size,
                              hipStream_t stream) {
    const float* targets = (const float*)d_in[0];
    const float* preds   = (const float*)d_in[1];

    unsigned long long total = (unsigned long long)in_sizes[0];   // B*N*13
    unsigned long long rows  = total / 13ull;                     // B*N
    unsigned int numGroups   = (unsigned int)(rows / 4ull);       // 4 rows/thread-unit

    float* partials = (float*)d_ws;   // NBLOCKS * 4 floats

    loss_partial_kernel<<<NBLOCKS, NTHREADS, 0, stream>>>(
        targets, preds, partials, numGroups);
    loss_final_kernel<<<1, 32, 0, stream>>>(
        partials, (float*)d_out, 1.0f / (float)rows, NBLOCKS);
}